// DCNv2Block_52845277610552
// MI455X (gfx1250) — compile-verified
//
#include <hip/hip_runtime.h>
#include <math.h>

#define B_    4
#define CIN_  256
#define COUT_ 256
#define H_    64
#define W_    64
#define KK_   9
#define HW_   (H_ * W_)          // 4096
#define KDIM_ (CIN_ * KK_)       // 2304
#define CCHUNK 64
#define KCHUNK (CCHUNK * KK_)    // 576
#define LROW   (KCHUNK + 4)      // 580 words: B-frag LDS reads bank-conflict-free
#define NPIX   32                // pixels per workgroup (2 N-tiles of 16)
#define NPAIR  (NPIX * KK_)      // 288 (pixel,tap) pairs

typedef __attribute__((ext_vector_type(2))) float v2f;
typedef __attribute__((ext_vector_type(8))) float v8f;

// ---------------------------------------------------------------------------
// Kernel 0: NCHW -> NHWC transpose of x (LDS tiled). 4096 blocks x 256 thr.
// ---------------------------------------------------------------------------
__global__ void __launch_bounds__(256)
k_transpose(const float* __restrict__ x, float* __restrict__ xT) {
    __shared__ float tile[16][W_ + 1];
    int blk = blockIdx.x;
    int cg  = blk & 15;
    int y   = (blk >> 4) & 63;
    int b   = blk >> 10;
    int t   = threadIdx.x;
    int tx  = t & 63;
    int tc  = t >> 6;            // 0..3
#pragma unroll
    for (int i = 0; i < 4; ++i) {
        int cl = tc * 4 + i;
        int c  = cg * 16 + cl;
        tile[cl][tx] = x[(((b * CIN_) + c) * H_ + y) * W_ + tx];
    }
    __syncthreads();
    int co    = t & 15;
    int xbase = (t >> 4) * 4;
#pragma unroll
    for (int j = 0; j < 4; ++j) {
        int xo = xbase + j;
        xT[(((b * H_) + y) * W_ + xo) * CIN_ + cg * 16 + co] = tile[co][xo];
    }
}

// ---------------------------------------------------------------------------
// Kernel 1: 27-channel offset conv (+bias) -> dy/dx + sigmoid mask.
// One thread per (b, co, pixel): 1728 blocks x 256 thr.
// ---------------------------------------------------------------------------
__global__ void __launch_bounds__(256)
k_offsets(const float* __restrict__ x, const float* __restrict__ ow,
          const float* __restrict__ ob,
          float* __restrict__ dyp, float* __restrict__ dxp,
          float* __restrict__ mkp) {
    int gid = blockIdx.x * 256 + threadIdx.x;
    int pix = gid & (HW_ - 1);
    int co  = (gid >> 12) % 27;
    int b   = gid / (27 * HW_);
    int y   = pix >> 6, xx = pix & 63;

    float acc = ob[co];
    const float* wrow = ow + co * KDIM_;
    for (int ci = 0; ci < CIN_; ++ci) {
        const float* xc = x + ((b * CIN_ + ci) * H_) * W_;
        const float* wc = wrow + ci * 9;
#pragma unroll
        for (int ki = 0; ki < 3; ++ki) {
            int  yy  = y - 1 + ki;
            bool yok = (yy >= 0) & (yy < H_);
#pragma unroll
            for (int kj = 0; kj < 3; ++kj) {
                int  xj = xx - 1 + kj;
                bool ok = yok & (xj >= 0) & (xj < W_);
                float xv = ok ? xc[yy * W_ + xj] : 0.f;
                acc = fmaf(xv, wc[ki * 3 + kj], acc);
            }
        }
    }
    if (co < 18) {
        int k = co >> 1;
        float* dst = (co & 1) ? dxp : dyp;
        dst[(b * KK_ + k) * HW_ + pix] = acc;
    } else {
        mkp[(b * KK_ + (co - 18)) * HW_ + pix] = 1.f / (1.f + __expf(-acc));
    }
}

// ---------------------------------------------------------------------------
// Kernel 2: fused bilinear-sample im2col + WMMA GEMM.
// Grid: B * 128 tiles (32 px each). Block: 256 thr = 8 waves.
// Each wave: 2 M-tiles x 2 N-tiles -> 4 accumulators; each A-fragment pair
// feeds 4 v_wmma_f32_16x16x4_f32. K = 2304 in 4 LDS chunks of 576.
// ---------------------------------------------------------------------------
__global__ void __launch_bounds__(256)
k_dcn_wmma(const float* __restrict__ xT, const float* __restrict__ wk,
           const float* __restrict__ dyp, const float* __restrict__ dxp,
           const float* __restrict__ mkp, float* __restrict__ out) {
    __shared__ float s_col[NPIX * LROW];                   // 74240 B im2col slab
    __shared__ float s_wy[NPAIR], s_wx[NPAIR], s_m[NPAIR];
    __shared__ int   s_y0[NPAIR], s_x0[NPAIR];

    int tile = blockIdx.x;        // 0..511
    int b    = tile >> 7;
    int pix0 = (tile & 127) * NPIX;
    int t    = threadIdx.x;

    // Per-(pixel,tap) bilinear parameters (computed once).
    for (int i = t; i < NPAIR; i += 256) {
        int p = i / 9, k = i % 9;
        int pix = pix0 + p;
        int y = pix >> 6, xx = pix & 63;
        int ki = k / 3, kj = k % 3;
        int oidx = (b * KK_ + k) * HW_ + pix;
        float py = (float)(y - 1 + ki) + dyp[oidx];
        float px = (float)(xx - 1 + kj) + dxp[oidx];
        float fy = floorf(py), fx = floorf(px);
        s_y0[i] = (int)fy;
        s_x0[i] = (int)fx;
        s_wy[i] = py - fy;
        s_wx[i] = px - fx;
        s_m[i]  = mkp[oidx];
    }

    int lane = t & 31;
    int wave = t >> 5;           // 0..7
    int half = lane >> 4;        // K-half select per WMMA f32 A/B layout
    int l15  = lane & 15;
    int m0a  = wave * 32;        // first M-tile base row
    int m0b  = wave * 32 + 16;   // second M-tile base row

    v8f accA0 = {0.f,0.f,0.f,0.f,0.f,0.f,0.f,0.f};
    v8f accA1 = accA0, accB0 = accA0, accB1 = accA0;

    int c2  = t & 31;            // float2 channel pair within chunk (64 ch)
    int grp = t >> 5;            // 0..7, each covers 36 (pixel,tap) pairs

    const float* wrowA = wk + (size_t)(m0a + l15) * KDIM_;
    const float* wrowB = wk + (size_t)(m0b + l15) * KDIM_;

    for (int chunk = 0; chunk < CIN_ / CCHUNK; ++chunk) {
        __syncthreads();         // params ready / prior GEMM LDS reads done
        int c0 = chunk * CCHUNK;

        // Prefetch this chunk's weight rows toward near caches while we build.
        __builtin_prefetch(wrowA + chunk * KCHUNK + 2 * half, 0, 0);
        __builtin_prefetch(wrowB + chunk * KCHUNK + 2 * half, 0, 0);

        // ---- build im2col chunk: s_col[p][ (c-c0)*9 + k ], float2 per thread
        const float* xb = xT + (size_t)(b * HW_) * CIN_ + c0 + 2 * c2;
        for (int pk = grp * 36; pk < grp * 36 + 36; ++pk) {
            int p  = pk / 9, k = pk % 9;
            int y0 = s_y0[pk], x0 = s_x0[pk];
            float wy = s_wy[pk], wx = s_wx[pk], m = s_m[pk];
            float w00 = (1.f - wy) * (1.f - wx) * m;
            float w01 = (1.f - wy) * wx * m;
            float w10 = wy * (1.f - wx) * m;
            float w11 = wy * wx * m;
            bool y0ok = (y0 >= 0) & (y0 < H_);
            bool y1ok = (y0 + 1 >= 0) & (y0 + 1 < H_);
            bool x0ok = (x0 >= 0) & (x0 < W_);
            bool x1ok = (x0 + 1 >= 0) & (x0 + 1 < W_);
            float v0 = 0.f, v1 = 0.f;
            if (y0ok & x0ok) {
                float2 g = *(const float2*)(xb + (size_t)(y0 * W_ + x0) * CIN_);
                v0 = fmaf(g.x, w00, v0); v1 = fmaf(g.y, w00, v1);
            }
            if (y0ok & x1ok) {
                float2 g = *(const float2*)(xb + (size_t)(y0 * W_ + x0 + 1) * CIN_);
                v0 = fmaf(g.x, w01, v0); v1 = fmaf(g.y, w01, v1);
            }
            if (y1ok & x0ok) {
                float2 g = *(const float2*)(xb + (size_t)((y0 + 1) * W_ + x0) * CIN_);
                v0 = fmaf(g.x, w10, v0); v1 = fmaf(g.y, w10, v1);
            }
            if (y1ok & x1ok) {
                float2 g = *(const float2*)(xb + (size_t)((y0 + 1) * W_ + x0 + 1) * CIN_);
                v0 = fmaf(g.x, w11, v0); v1 = fmaf(g.y, w11, v1);
            }
            s_col[p * LROW + (2 * c2) * KK_ + k]     = v0;
            s_col[p * LROW + (2 * c2 + 1) * KK_ + k] = v1;
        }
        __syncthreads();

        // ---- GEMM over this K-chunk: 144 k-steps, 4 WMMA per step ----
        for (int kk = 0; kk < KCHUNK; kk += 4) {
            int kb = kk + 2 * half;      // lanes 0-15: K+0/1, lanes 16-31: K+2/3
            v2f b0, b1, aA, aB;
            b0.x = s_col[l15 * LROW + kb];
            b0.y = s_col[l15 * LROW + kb + 1];
            b1.x = s_col[(16 + l15) * LROW + kb];
            b1.y = s_col[(16 + l15) * LROW + kb + 1];
            int gK = chunk * KCHUNK + kb;
            aA.x = wrowA[gK]; aA.y = wrowA[gK + 1];
            aB.x = wrowB[gK]; aB.y = wrowB[gK + 1];
            accA0 = __builtin_amdgcn_wmma_f32_16x16x4_f32(
                false, aA, false, b0, (short)0, accA0, false, false);
            accA1 = __builtin_amdgcn_wmma_f32_16x16x4_f32(
                false, aA, false, b1, (short)0, accA1, false, false);
            accB0 = __builtin_amdgcn_wmma_f32_16x16x4_f32(
                false, aB, false, b0, (short)0, accB0, false, false);
            accB1 = __builtin_amdgcn_wmma_f32_16x16x4_f32(
                false, aB, false, b1, (short)0, accB1, false, false);
        }
    }

    // ---- store D (16x16 f32 layout: VGPR r -> M = r + 8*half, N = l15) ----
#pragma unroll
    for (int r = 0; r < 8; ++r) {
        int mA = m0a + r + 8 * half;
        int mB = m0b + r + 8 * half;
        size_t rowA = ((size_t)(b * COUT_ + mA)) * HW_ + pix0;
        size_t rowB = ((size_t)(b * COUT_ + mB)) * HW_ + pix0;
        out[rowA + l15]      = accA0[r];
        out[rowA + 16 + l15] = accA1[r];
        out[rowB + l15]      = accB0[r];
        out[rowB + 16 + l15] = accB1[r];
    }
}

// ---------------------------------------------------------------------------
extern "C" void kernel_launch(void* const* d_in, const int* in_sizes, int n_in,
                              void* d_out, int out_size, void* d_ws, size_t ws_size,
                              hipStream_t stream) {
    const float* x    = (const float*)d_in[0];   // (4,256,64,64)
    const float* ow   = (const float*)d_in[1];   // (27,256,3,3)
    const float* ob   = (const float*)d_in[2];   // (27,)
    const float* dcnw = (const float*)d_in[3];   // (256,256,3,3)
    float* out = (float*)d_out;                  // (4,256,64,64)

    float* xT  = (float*)d_ws;                               // 4*64*64*256
    float* dyp = xT  + (size_t)B_ * H_ * W_ * CIN_;          // 4*9*4096
    float* dxp = dyp + (size_t)B_ * KK_ * HW_;
    float* mkp = dxp + (size_t)B_ * KK_ * HW_;

    k_transpose<<<B_ * H_ * (CIN_ / 16), 256, 0, stream>>>(x, xT);
    k_offsets  <<<(B_ * 27 * HW_) / 256, 256, 0, stream>>>(x, ow, ob, dyp, dxp, mkp);
    k_dcn_wmma <<<B_ * (HW_ / NPIX),     256, 0, stream>>>(xT, dcnw, dyp, dxp, mkp, out);
}